// adder2d_qp_81166291960325
// MI455X (gfx1250) — compile-verified
//
#include <hip/hip_runtime.h>
#include <hip/hip_bf16.h>

// AdderNet L1-distance "convolution":
// out[n,co,ho,wo] = -sum_{ci,kh,kw} mask[co,d] * |W[co,d] - xpad[n,ci,ho+kh-1,wo+kw-1]|
// N=4, Cin=64, H=W=32, Cout=64, k=3, stride=1, pad=1, Ho=Wo=32.

#define NB    4
#define CIN   64
#define HH    32
#define WW    32
#define COUT  64
#define KS    3
#define HO    32
#define WO    32
#define DD    (CIN*KS*KS)            // 576

#define TILE_R     4                 // output rows per block
#define ROW_TILES  (HO/TILE_R)       // 8
#define COUT_TILE  16
#define COUT_TILES (COUT/COUT_TILE)  // 4
#define XROWS      (TILE_R+2)        // 6 input rows incl. halo
#define XSTRIDE    40                // floats/row: col c -> idx c+4; halos at idx 3 and 36 (16B-aligned interior)
#define THREADS    256

#define WM_FLOATS  (COUT_TILE*DD*2)        // 18432 floats (W,mask interleaved)
#define XS_FLOATS  (CIN*XROWS*XSTRIDE)     // 15360 floats
#define SMEM_BYTES ((WM_FLOATS + XS_FLOATS) * 4)   // 135168 B < 320KB/WGP

#if defined(__gfx1250__) && __has_builtin(__builtin_amdgcn_global_load_async_to_lds_b128)
#define USE_ASYNC_LDS 1
#else
#define USE_ASYNC_LDS 0
#endif

typedef int v4i_t __attribute__((ext_vector_type(4)));
typedef __attribute__((address_space(1))) v4i_t* gv4i_p;   // global (__device__) v4i*
typedef __attribute__((address_space(3))) v4i_t* lv4i_p;   // LDS (__shared__) v4i*

__device__ __forceinline__ void copy16_to_lds(float* lds_dst, const float* gsrc) {
#if USE_ASYNC_LDS
    // CDNA5 async copy: global -> LDS, tracked by ASYNCcnt (no VGPR data path).
    __builtin_amdgcn_global_load_async_to_lds_b128(
        (gv4i_p)gsrc,      // C-style cast: drops const, addrspacecast generic->AS1
        (lv4i_p)lds_dst,   // addrspacecast generic->AS3
        /*offset=*/0, /*cpol=*/0);
#else
    *(float4*)lds_dst = *(const float4*)gsrc;
#endif
}

__device__ __forceinline__ void wait_async_done() {
#if USE_ASYNC_LDS
#if __has_builtin(__builtin_amdgcn_s_wait_asynccnt)
    __builtin_amdgcn_s_wait_asynccnt(0);
#else
    asm volatile("s_wait_asynccnt 0" ::: "memory");
#endif
#endif
}

extern "C" __global__ __launch_bounds__(THREADS)
void adder2d_qp_81166291960325_kernel(const float* __restrict__ xg,
                                      const float* __restrict__ wg,
                                      const float* __restrict__ mg,
                                      float* __restrict__ outg) {
    extern __shared__ __align__(16) float smem[];
    float2* __restrict__ wm2 = (float2*)smem;          // [COUT_TILE * DD] (w,m) pairs
    float*  __restrict__ xs  = smem + WM_FLOATS;       // [CIN][XROWS][XSTRIDE]

    const int tid = threadIdx.x;
    int b = blockIdx.x;
    const int ct = b & (COUT_TILES - 1); b >>= 2;      // Cout tile
    const int rt = b & (ROW_TILES - 1);  b >>= 3;      // row tile
    const int n  = b;                                  // image

    // ---- stage W & mask interleaved (broadcast-read layout) ----
    {
        const int base = ct * COUT_TILE * DD;          // contiguous co-rows in W_col/mask
        for (int i = tid; i < COUT_TILE * DD; i += THREADS) {
            float2 v;
            v.x = wg[base + i];
            v.y = mg[base + i];
            wm2[i] = v;
        }
    }

    // ---- zero x slab (provides padding halo + out-of-range rows) ----
    for (int i = tid; i < XS_FLOATS; i += THREADS) xs[i] = 0.0f;
    __syncthreads();

    // ---- async-stage interior x rows: 64 ci * 6 rows * 8 x 16B segments ----
    {
        const int r0 = rt * TILE_R - 1;
        const float* xn = xg + (size_t)n * CIN * HH * WW;
        for (int idx = tid; idx < CIN * XROWS * (WW / 4); idx += THREADS) {
            const int seg = idx & 7;                   // 16B segment within row
            const int row = idx >> 3;
            const int ci  = row / XROWS;
            const int lr  = row - ci * XROWS;
            const int r   = r0 + lr;
            if (r >= 0 && r < HH) {
                const float* src = xn + (ci * HH + r) * WW + seg * 4;
                float* dst = xs + (ci * XROWS + lr) * XSTRIDE + 4 + seg * 4;  // 16B aligned
                copy16_to_lds(dst, src);
            }
        }
    }
    wait_async_done();
    __syncthreads();

    // ---- compute: 2 pixels x 4 couts per thread ----
    const int pixslot = tid & 63;
    const int cog     = tid >> 6;        // 0..3, wave-uniform -> wm reads are LDS broadcasts
    const int prp     = pixslot >> 5;    // row-pair within tile (wave-uniform)
    const int pcol    = pixslot & 31;    // lane-consecutive -> conflict-free xs reads

    float acc[8];
#pragma unroll
    for (int i = 0; i < 8; ++i) acc[i] = 0.0f;

    const float2* __restrict__ wmt = wm2 + (size_t)cog * 4 * DD;  // local co = cog*4 + j

    for (int ci = 0; ci < CIN; ++ci) {
        const float* xci = xs + ci * (XROWS * XSTRIDE) + prp * 2 * XSTRIDE + 3 + pcol;
        const int dci = ci * 9;
#pragma unroll
        for (int kh = 0; kh < KS; ++kh) {
#pragma unroll
            for (int kw = 0; kw < KS; ++kw) {
                const float xv0 = xci[(kh + 0) * XSTRIDE + kw];   // pixel row prp*2+0
                const float xv1 = xci[(kh + 1) * XSTRIDE + kw];   // pixel row prp*2+1
                const int d = dci + kh * 3 + kw;
#pragma unroll
                for (int j = 0; j < 4; ++j) {
                    const float2 wmv = wmt[j * DD + d];           // ds_load_b64 broadcast
                    acc[j]     = fmaf(wmv.y, fabsf(wmv.x - xv0), acc[j]);
                    acc[4 + j] = fmaf(wmv.y, fabsf(wmv.x - xv1), acc[4 + j]);
                }
            }
        }
    }

    // ---- write out[n][co][ho][wo], negated; coalesced across lanes ----
#pragma unroll
    for (int pp = 0; pp < 2; ++pp) {
        const int ho = rt * TILE_R + prp * 2 + pp;
#pragma unroll
        for (int j = 0; j < 4; ++j) {
            const int co = ct * COUT_TILE + cog * 4 + j;
            outg[(((size_t)n * COUT + co) * HO + ho) * WO + pcol] = -acc[pp * 4 + j];
        }
    }
}

extern "C" void kernel_launch(void* const* d_in, const int* in_sizes, int n_in,
                              void* d_out, int out_size, void* d_ws, size_t ws_size,
                              hipStream_t stream) {
    const float* x = (const float*)d_in[0];   // (4,64,32,32)
    const float* W = (const float*)d_in[1];   // (64,64,3,3) == W_col (64,576)
    const float* M = (const float*)d_in[2];   // (64,576)
    float* out = (float*)d_out;               // (4,64,32,32)

    dim3 grid(NB * ROW_TILES * COUT_TILES);   // 128 blocks
    dim3 block(THREADS);                      // 8 waves (wave32)
    adder2d_qp_81166291960325_kernel<<<grid, block, SMEM_BYTES, stream>>>(x, W, M, out);
}